// Decoder_63591285784721
// MI455X (gfx1250) — compile-verified
//
#include <hip/hip_runtime.h>
#include <math.h>

#define VOCAB 5
#define DDIM  64
#define BATCH 8
#define SEQ   2048
#define TQ    256

typedef __attribute__((ext_vector_type(2))) float v2f;
typedef __attribute__((ext_vector_type(8))) float v8f;

// ---------------------------------------------------------------------------
// Kernel 1 (one wave32): build Qtab/Ktab/Vtab[5][64] = emb_pad(16x64) @ W.T + b
// via V_WMMA_F32_16X16X4_F32, then Sdot[5][5] = (Qtab @ Ktab.T) / sqrt(64).
// ---------------------------------------------------------------------------
__global__ __launch_bounds__(32) void qkv_tables_kernel(
    const float* __restrict__ emb,
    const float* __restrict__ wq, const float* __restrict__ bq,
    const float* __restrict__ wk, const float* __restrict__ bk,
    const float* __restrict__ wv, const float* __restrict__ bv,
    float* __restrict__ sdot_out, float* __restrict__ vtab_out)
{
    __shared__ float sE[16 * DDIM];     // zero-padded embedding rows 5..15
    __shared__ float sW[DDIM * DDIM];   // current weight matrix
    __shared__ float sQ[VOCAB * DDIM];
    __shared__ float sK[VOCAB * DDIM];

    const int tid    = threadIdx.x;
    const int half   = tid >> 4;        // 0: lanes 0-15, 1: lanes 16-31
    const int lane16 = tid & 15;

    for (int i = tid; i < 16 * DDIM; i += 32)
        sE[i] = (i < VOCAB * DDIM) ? emb[i] : 0.0f;

    const float* Ws[3] = {wq, wk, wv};
    const float* Bs[3] = {bq, bk, bv};

    #pragma unroll 1
    for (int p = 0; p < 3; ++p) {
        __syncthreads();
        for (int i = tid; i < DDIM * DDIM; i += 32) sW[i] = Ws[p][i];
        __syncthreads();
        const float* bias = Bs[p];

        for (int nt = 0; nt < 4; ++nt) {      // 4 tiles of 16 output columns
            v8f acc = {};
            #pragma unroll
            for (int kk = 0; kk < 16; ++kk) { // K = 64 in steps of 4
                const int k0 = kk * 4 + half * 2;
                v2f a, b;
                // A 16x4 f32 frag: lanes 0-15 hold K=k0,k0+1; lanes 16-31 K=k0+2,k0+3
                a.x = sE[lane16 * DDIM + k0];
                a.y = sE[lane16 * DDIM + k0 + 1];
                // B 4x16 frag: B[k][n] = W[n][k]  (y = x @ W.T)
                const int n = nt * 16 + lane16;
                b.x = sW[n * DDIM + k0];
                b.y = sW[n * DDIM + k0 + 1];
                acc = __builtin_amdgcn_wmma_f32_16x16x4_f32(
                    false, a, false, b, (short)0, acc, false, false);
            }
            // C/D layout: VGPR r, lanes 0-15 -> row M=r (rows 0..4 are real)
            if (half == 0) {
                const int n  = nt * 16 + lane16;
                const float bn = bias[n];
                #pragma unroll
                for (int r = 0; r < VOCAB; ++r) {
                    const float val = acc[r] + bn;
                    if (p == 0)      sQ[r * DDIM + n] = val;
                    else if (p == 1) sK[r * DDIM + n] = val;
                    else             vtab_out[r * DDIM + n] = val;
                }
            }
        }
    }
    __syncthreads();

    // 5x5 score table, scaled by 1/sqrt(D) = 0.125
    if (tid < VOCAB * VOCAB) {
        const int tq = tid / VOCAB, tk = tid % VOCAB;
        float d = 0.0f;
        for (int c = 0; c < DDIM; ++c) d += sQ[tq * DDIM + c] * sK[tk * DDIM + c];
        sdot_out[tid] = d * 0.125f;
    }
}

// ---------------------------------------------------------------------------
// Kernel 2: causal attention via prefix vocab counts.
// Grid (SEQ/TQ, BATCH), 256 threads. Each thread owns one query, then a
// cooperative coalesced write phase (16 lanes per 256B output row).
// ---------------------------------------------------------------------------
__global__ __launch_bounds__(TQ) void attn_kernel(
    const int*   __restrict__ x,
    const float* __restrict__ sdot,
    const float* __restrict__ vtab,
    float*       __restrict__ out)
{
    __shared__ int   sTok[TQ];
    __shared__ float sS[VOCAB * VOCAB];
    __shared__ float sV[VOCAB * DDIM];
    __shared__ int   sCnt[VOCAB];
    __shared__ float sWn[TQ * VOCAB];

    const int tid = threadIdx.x;
    const int b   = blockIdx.y;
    const int q0  = blockIdx.x * TQ;
    const int* xb = x + b * SEQ;

    __builtin_prefetch(xb + q0, 0, 1);  // global_prefetch_b8

    if (tid < VOCAB * VOCAB) sS[tid] = sdot[tid];
    for (int i = tid; i < VOCAB * DDIM; i += TQ) sV[i] = vtab[i];
    if (tid < VOCAB) sCnt[tid] = 0;
    sTok[tid] = xb[q0 + tid];
    __syncthreads();

    // base prefix counts for tokens [0, q0)
    int c[VOCAB] = {0, 0, 0, 0, 0};
    for (int j = tid; j < q0; j += TQ) {
        const int t = xb[j];
        #pragma unroll
        for (int v = 0; v < VOCAB; ++v) c[v] += (t == v);
    }
    #pragma unroll
    for (int v = 0; v < VOCAB; ++v)
        if (c[v]) atomicAdd(&sCnt[v], c[v]);
    __syncthreads();

    // per-thread in-tile scan: counts over keys k <= q  (q = q0 + tid)
    int cnt[VOCAB];
    #pragma unroll
    for (int v = 0; v < VOCAB; ++v) cnt[v] = sCnt[v];
    for (int j = 0; j <= tid; ++j) {
        const int t = sTok[j];
        #pragma unroll
        for (int v = 0; v < VOCAB; ++v) cnt[v] += (t == v);
    }

    // grouped softmax: w_v = cnt_v * exp(s_v - m), m = max over present ids
    const int tq = sTok[tid];
    float m = -INFINITY;
    #pragma unroll
    for (int v = 0; v < VOCAB; ++v)
        if (cnt[v] > 0) m = fmaxf(m, sS[tq * VOCAB + v]);
    float w[VOCAB], sumw = 0.0f;
    #pragma unroll
    for (int v = 0; v < VOCAB; ++v) {
        const float e = (cnt[v] > 0)
            ? (float)cnt[v] * __expf(sS[tq * VOCAB + v] - m) : 0.0f;
        w[v] = e; sumw += e;
    }
    const float inv = 1.0f / sumw;
    #pragma unroll
    for (int v = 0; v < VOCAB; ++v) sWn[tid * VOCAB + v] = w[v] * inv;
    __syncthreads();

    // cooperative coalesced store: 16 lanes write one query row (256 B)
    const float4* sV4  = (const float4*)sV;
    float4*       out4 = (float4*)(out + (size_t)(b * SEQ + q0) * DDIM);
    const int col  = tid & 15;   // float4 column within row
    const int qsub = tid >> 4;   // query sub-index within chunk
    for (int chunk = 0; chunk < 16; ++chunk) {
        const int qq = chunk * 16 + qsub;
        const float wn0 = sWn[qq * VOCAB + 0];
        const float wn1 = sWn[qq * VOCAB + 1];
        const float wn2 = sWn[qq * VOCAB + 2];
        const float wn3 = sWn[qq * VOCAB + 3];
        const float wn4 = sWn[qq * VOCAB + 4];
        const float4 a0 = sV4[0 * 16 + col];
        const float4 a1 = sV4[1 * 16 + col];
        const float4 a2 = sV4[2 * 16 + col];
        const float4 a3 = sV4[3 * 16 + col];
        const float4 a4 = sV4[4 * 16 + col];
        float4 r;
        r.x = wn0*a0.x + wn1*a1.x + wn2*a2.x + wn3*a3.x + wn4*a4.x;
        r.y = wn0*a0.y + wn1*a1.y + wn2*a2.y + wn3*a3.y + wn4*a4.y;
        r.z = wn0*a0.z + wn1*a1.z + wn2*a2.z + wn3*a3.z + wn4*a4.z;
        r.w = wn0*a0.w + wn1*a1.w + wn2*a2.w + wn3*a3.w + wn4*a4.w;
        out4[qq * 16 + col] = r;
    }
}

extern "C" void kernel_launch(void* const* d_in, const int* in_sizes, int n_in,
                              void* d_out, int out_size, void* d_ws, size_t ws_size,
                              hipStream_t stream) {
    const int*   x   = (const int*)d_in[0];
    const float* emb = (const float*)d_in[1];
    const float* wq  = (const float*)d_in[2];
    const float* bq  = (const float*)d_in[3];
    const float* wk  = (const float*)d_in[4];
    const float* bk  = (const float*)d_in[5];
    const float* wv  = (const float*)d_in[6];
    const float* bv  = (const float*)d_in[7];

    float* sdot = (float*)d_ws;        // 25 floats (padded to 32)
    float* vtab = (float*)d_ws + 32;   // 5*64 floats

    qkv_tables_kernel<<<1, 32, 0, stream>>>(emb, wq, bq, wk, bk, wv, bv,
                                            sdot, vtab);
    dim3 grid(SEQ / TQ, BATCH);
    attn_kernel<<<grid, TQ, 0, stream>>>(x, sdot, vtab, (float*)d_out);
}